// EHIAAttention_7456063226172
// MI455X (gfx1250) — compile-verified
//
#include <hip/hip_runtime.h>
#include <math.h>

// ---------------------------------------------------------------------------
// EHIA attention, MI455X (gfx1250, wave32, WMMA f32 16x16x4).
// Memory-bound: ~232 GFLOP fp32 vs ~604 MB mandatory HBM traffic (x read
// twice + out once) => ~26 us @ 23.3 TB/s.  Design choices:
//  * q is never materialized (only its sequence reductions are needed),
//  * k is recomputed in the output pass instead of spilling 201 MB,
//  * V_WMMA_F32_16X16X4_F32 keeps full fp32 precision (compute is free here),
//  * 2x2 register blocking per wave: A frags reused across 2 column tiles,
//    B frags across 2 row groups -> 1 weight dword per WMMA instead of 2,
//  * weights repacked once into K-packs-of-4 so every B fragment is one
//    aligned global_load_b64 (cuts hot-loop VMEM instruction count 2x and
//    keeps weight L2 traffic ~= weight_size per block).
// ---------------------------------------------------------------------------

#define NPATCH 4096
#define DIM    768
#define NHEADS 12
#define INDIMS 64
#define MT     32            // rows (patches) per block tile
#define LDPAD  772           // padded LDS row stride (breaks bank conflicts)
#define SCALE_ 0.015625f     // 4096^-0.5
#define LN_EPS 1e-5f

typedef float v2f __attribute__((ext_vector_type(2)));
typedef float v8f __attribute__((ext_vector_type(8)));

#define WMMA_F32(A, B, C) \
    __builtin_amdgcn_wmma_f32_16x16x4_f32(false, (A), false, (B), (short)0, (C), false, false)

// 16x16x4 f32 WMMA fragment indices (wave32):
//   A (16x4, MxK):  lane l, vgpr v -> M = l&15,  K = v + 2*(l>>4)
//   B (4x16, KxN):  lane l, vgpr v -> N = l&15,  K = v + 2*(l>>4)
//   C/D (16x16):    lane l, vgpr r -> N = l&15,  M = r + 8*(l>>4)
//
// Packed weight layout: wp[((k>>2)*DIM + c)*4 + (k&3)]  ->  the two K values
// a lane needs (k = kk+2*lhi, kk+2*lhi+1) are adjacent: one b64 load.

// ---------------------------------------------------------------------------
// Kernel P: repack a 768x768 row-major weight into K-packs-of-4.
// ---------------------------------------------------------------------------
__global__ __launch_bounds__(256)
void ehia_repack(const float* __restrict__ w, float* __restrict__ wp)
{
    const int i = blockIdx.x * 256 + threadIdx.x;     // grid == DIM*DIM/256
    const int k = i / DIM, c = i - k * DIM;
    wp[((size_t)(k >> 2) * DIM + c) * 4 + (k & 3)] = w[i];
}

// ---------------------------------------------------------------------------
// Kernel A: q/k tile GEMMs (2x2-blocked WMMA) folded straight into the three
// sequence reductions qw, qsum, ksum via shuffles + global float atomics.
// ---------------------------------------------------------------------------
__device__ __forceinline__ void fold_reduce(
    const v8f& q0, const v8f& q1, const v8f& k0, const v8f& k1,
    int c, int b, int m0, int lhi,
    const float* __restrict__ bq, const float* __restrict__ bk,
    const float* __restrict__ w_g,
    float* __restrict__ qw_buf, float* __restrict__ qs_buf,
    float* __restrict__ ks_buf)
{
    const float bqc = bq[c], bkc = bk[c];
    float lqw = 0.f, lqs = 0.f, lks = 0.f;
#pragma unroll
    for (int r = 0; r < 8; ++r) {
        const int r0 = r + 8 * lhi, r1 = 16 + r0;     // rows in block tile
        const float w0 = w_g[m0 + r0], w1 = w_g[m0 + r1];
        const float q0v = q0[r] + bqc, q1v = q1[r] + bqc;
        const float k0v = k0[r] + bkc, k1v = k1[r] + bkc;
        lqw += w0 * q0v + w1 * q1v;
        lqs += q0v + q1v;
        lks += k0v + k1v;
    }
    lqw += __shfl_xor(lqw, 16);
    lqs += __shfl_xor(lqs, 16);
    lks += __shfl_xor(lks, 16);
    if (lhi == 0) {
        atomicAdd(&qw_buf[b * DIM + c], lqw);
        atomicAdd(&qs_buf[b * DIM + c], lqs);
        atomicAdd(&ks_buf[b * DIM + c], lks);
    }
}

__global__ __launch_bounds__(256)
void ehia_qk_reduce(const float* __restrict__ x,
                    const float* __restrict__ wqp, const float* __restrict__ bq,
                    const float* __restrict__ wkp, const float* __restrict__ bk,
                    const float* __restrict__ w_g,
                    float* __restrict__ qw_buf,
                    float* __restrict__ qs_buf,
                    float* __restrict__ ks_buf)
{
    extern __shared__ __align__(16) float smem[];
    float* x_s = smem;                               // MT x LDPAD
    const int b   = blockIdx.y;
    const int m0  = blockIdx.x * MT;
    const int tid = threadIdx.x;

    const float4* xg = (const float4*)(x + ((size_t)b * NPATCH + m0) * DIM);
    for (int v = tid; v < MT * DIM / 4; v += 256) {
        int f = v * 4, row = f / DIM, col = f % DIM;
        *(float4*)&x_s[row * LDPAD + col] = xg[(row * DIM + col) >> 2];
    }
    __syncthreads();

    const int wv  = tid >> 5, ln = tid & 31;
    const int cg  = wv;                  // column group: 96 cols = 3 tile pairs
    const int lhi = ln >> 4, llo = ln & 15;

    for (int tp = 0; tp < 3; ++tp) {
        const int n0 = cg * 96 + tp * 32;
        const int ca = n0 + llo, cb = n0 + 16 + llo;
        v8f q00 = {0,0,0,0,0,0,0,0}, q10 = {0,0,0,0,0,0,0,0};
        v8f q01 = {0,0,0,0,0,0,0,0}, q11 = {0,0,0,0,0,0,0,0};
        v8f k00 = {0,0,0,0,0,0,0,0}, k10 = {0,0,0,0,0,0,0,0};
        v8f k01 = {0,0,0,0,0,0,0,0}, k11 = {0,0,0,0,0,0,0,0};
        for (int kk = 0; kk < DIM; kk += 4) {
            const int kb = kk + 2 * lhi;
            const v2f a0 = *(const v2f*)&x_s[llo * LDPAD + kb];          // rows 0..15
            const v2f a1 = *(const v2f*)&x_s[(16 + llo) * LDPAD + kb];   // rows 16..31
            const size_t wb = (size_t)(kk >> 2) * (DIM * 4) + 2 * lhi;
            const v2f bqa = *(const v2f*)&wqp[wb + (size_t)ca * 4];
            const v2f bqb = *(const v2f*)&wqp[wb + (size_t)cb * 4];
            const v2f bka = *(const v2f*)&wkp[wb + (size_t)ca * 4];
            const v2f bkb = *(const v2f*)&wkp[wb + (size_t)cb * 4];
            q00 = WMMA_F32(a0, bqa, q00);  q10 = WMMA_F32(a1, bqa, q10);
            q01 = WMMA_F32(a0, bqb, q01);  q11 = WMMA_F32(a1, bqb, q11);
            k00 = WMMA_F32(a0, bka, k00);  k10 = WMMA_F32(a1, bka, k10);
            k01 = WMMA_F32(a0, bkb, k01);  k11 = WMMA_F32(a1, bkb, k11);
        }
        fold_reduce(q00, q10, k00, k10, ca, b, m0, lhi, bq, bk, w_g,
                    qw_buf, qs_buf, ks_buf);
        fold_reduce(q01, q11, k01, k11, cb, b, m0, lhi, bq, bk, w_g,
                    qw_buf, qs_buf, ks_buf);
    }
}

// ---------------------------------------------------------------------------
// Kernel B: per-batch gating chain -> g[b, d].  Tiny (16 blocks).
// mean over heads of (dv2 @ w_fc2 + b) == dv2 . colmean(w_fc2) + mean(b).
// ---------------------------------------------------------------------------
__global__ __launch_bounds__(256)
void ehia_gate(const float* __restrict__ qw_buf,
               const float* __restrict__ qs_buf,
               const float* __restrict__ ks_buf,
               const float* __restrict__ w_fc,  const float* __restrict__ b_fc,
               const float* __restrict__ w_fc2, const float* __restrict__ b_fc2,
               float* __restrict__ g_out)
{
    __shared__ float a_s[DIM];
    __shared__ float qm_s[DIM];
    __shared__ float hid[NHEADS * DIM];
    __shared__ float red[256];
    __shared__ float wbar[NHEADS];
    __shared__ float bbar_s;
    const int b = blockIdx.x, t = threadIdx.x;

    float loc = -3.4e38f;
    for (int i = t; i < DIM; i += 256) {
        float v = qw_buf[b * DIM + i] * SCALE_;
        a_s[i] = v; loc = fmaxf(loc, v);
    }
    red[t] = loc; __syncthreads();
    for (int s = 128; s > 0; s >>= 1) {
        if (t < s) red[t] = fmaxf(red[t], red[t + s]);
        __syncthreads();
    }
    const float mx = red[0]; __syncthreads();
    loc = 0.f;
    for (int i = t; i < DIM; i += 256) {
        float e = __expf(a_s[i] - mx); a_s[i] = e; loc += e;
    }
    red[t] = loc; __syncthreads();
    for (int s = 128; s > 0; s >>= 1) {
        if (t < s) red[t] += red[t + s];
        __syncthreads();
    }
    const float inv = 1.f / red[0]; __syncthreads();

    for (int i = t; i < DIM; i += 256) {
        a_s[i] *= inv;
        qm_s[i] = qs_buf[b * DIM + i] * (1.f / (float)NPATCH);
    }
    if (t < NHEADS) {
        float s = 0.f;
        for (int o = 0; o < NHEADS; ++o) s += w_fc2[t * NHEADS + o];
        wbar[t] = s * (1.f / NHEADS);
    }
    if (t == NHEADS) {
        float s = 0.f;
        for (int o = 0; o < NHEADS; ++o) s += b_fc2[o];
        bbar_s = s * (1.f / NHEADS);
    }
    __syncthreads();

    for (int o = t; o < NHEADS * DIM; o += 256) {
        const int h = o / DIM, j = o - h * DIM;
        float s = b_fc[j];
        for (int i = 0; i < INDIMS; ++i) s += qm_s[h * INDIMS + i] * w_fc[i * DIM + j];
        hid[o] = 0.5f * s * (1.f + erff(s * 0.70710678f));   // exact gelu
    }
    __syncthreads();

    const float bbar = bbar_s;
    for (int p = t; p < DIM; p += 256) {
        float s2 = bbar;                          // reshape (12,768)->(768,12)
        for (int h = 0; h < NHEADS; ++h) s2 += hid[p * NHEADS + h] * wbar[h];
        const float dvv = 1.f / (1.f + __expf(-s2));
        g_out[b * DIM + p] = a_s[p] * ks_buf[b * DIM + p] * dvv;
    }
}

// ---------------------------------------------------------------------------
// Kernel C: recompute k tile (2x2-blocked WMMA) into LDS, per-row LN stats,
// then (g .* k) @ w_proj (WMMA) with fused bias + LayerNorm epilogue.
// ---------------------------------------------------------------------------
__global__ __launch_bounds__(256)
void ehia_out(const float* __restrict__ x,
              const float* __restrict__ wkp, const float* __restrict__ bk,
              const float* __restrict__ g_buf,
              const float* __restrict__ wpp, const float* __restrict__ b_proj,
              const float* __restrict__ gamma,  const float* __restrict__ beta,
              float* __restrict__ out)
{
    extern __shared__ __align__(16) float smem[];
    float* x_s = smem;                    // MT x LDPAD (reused as scratch later)
    float* k_s = x_s + MT * LDPAD;        // MT x LDPAD
    float* g_s = k_s + MT * LDPAD;        // DIM
    float* mu  = g_s + DIM;               // MT
    float* rs  = mu + MT;                 // MT

    const int b   = blockIdx.y;
    const int m0  = blockIdx.x * MT;
    const int tid = threadIdx.x;

    const float4* xg = (const float4*)(x + ((size_t)b * NPATCH + m0) * DIM);
    for (int v = tid; v < MT * DIM / 4; v += 256) {
        int f = v * 4, row = f / DIM, col = f % DIM;
        *(float4*)&x_s[row * LDPAD + col] = xg[(row * DIM + col) >> 2];
    }
    for (int i = tid; i < DIM; i += 256) g_s[i] = g_buf[b * DIM + i];
    __syncthreads();

    const int wv  = tid >> 5, ln = tid & 31;
    const int cg  = wv;
    const int lhi = ln >> 4, llo = ln & 15;

    // ---- phase 1: k tile = x @ wk + bk  -> LDS ----
    for (int tp = 0; tp < 3; ++tp) {
        const int n0 = cg * 96 + tp * 32;
        const int ca = n0 + llo, cb = n0 + 16 + llo;
        v8f k00 = {0,0,0,0,0,0,0,0}, k10 = {0,0,0,0,0,0,0,0};
        v8f k01 = {0,0,0,0,0,0,0,0}, k11 = {0,0,0,0,0,0,0,0};
        for (int kk = 0; kk < DIM; kk += 4) {
            const int kb = kk + 2 * lhi;
            const v2f a0 = *(const v2f*)&x_s[llo * LDPAD + kb];
            const v2f a1 = *(const v2f*)&x_s[(16 + llo) * LDPAD + kb];
            const size_t wb = (size_t)(kk >> 2) * (DIM * 4) + 2 * lhi;
            const v2f ba = *(const v2f*)&wkp[wb + (size_t)ca * 4];
            const v2f bb = *(const v2f*)&wkp[wb + (size_t)cb * 4];
            k00 = WMMA_F32(a0, ba, k00);  k10 = WMMA_F32(a1, ba, k10);
            k01 = WMMA_F32(a0, bb, k01);  k11 = WMMA_F32(a1, bb, k11);
        }
        const float bka = bk[ca], bkb = bk[cb];
#pragma unroll
        for (int r = 0; r < 8; ++r) {
            const int r0 = r + 8 * lhi, r1 = 16 + r0;
            k_s[r0 * LDPAD + ca] = k00[r] + bka;
            k_s[r1 * LDPAD + ca] = k10[r] + bka;
            k_s[r0 * LDPAD + cb] = k01[r] + bkb;
            k_s[r1 * LDPAD + cb] = k11[r] + bkb;
        }
    }
    __syncthreads();

    // ---- phase 2: per-row LayerNorm statistics (x_s is dead -> scratch) ----
    {
        float* ps  = x_s;          // 256 partial sums
        float* ps2 = x_s + 256;    // 256 partial sumsq
        const int row = tid & 31, seg = tid >> 5;     // 8 segments of 96 cols
        float s = 0.f, s2 = 0.f;
        for (int j = seg * 96; j < seg * 96 + 96; ++j) {
            const float v = k_s[row * LDPAD + j];
            s += v; s2 += v * v;
        }
        ps[tid] = s; ps2[tid] = s2;
        __syncthreads();
        if (tid < MT) {
            float ts = 0.f, ts2 = 0.f;
            for (int g = 0; g < 8; ++g) { ts += ps[g * 32 + tid]; ts2 += ps2[g * 32 + tid]; }
            const float m = ts * (1.f / DIM);
            mu[tid] = m;
            rs[tid] = rsqrtf(fmaxf(ts2 * (1.f / DIM) - m * m, 0.f) + LN_EPS);
        }
        __syncthreads();
    }

    // ---- phase 3: (g .* k) @ w_proj + b_proj + LN(k) ----
    for (int tp = 0; tp < 3; ++tp) {
        const int n0 = cg * 96 + tp * 32;
        const int ca = n0 + llo, cb = n0 + 16 + llo;
        v8f o00 = {0,0,0,0,0,0,0,0}, o10 = {0,0,0,0,0,0,0,0};
        v8f o01 = {0,0,0,0,0,0,0,0}, o11 = {0,0,0,0,0,0,0,0};
        for (int kk = 0; kk < DIM; kk += 4) {
            const int kb = kk + 2 * lhi;
            v2f a0, a1;
            a0[0] = g_s[kb]     * k_s[llo * LDPAD + kb];
            a0[1] = g_s[kb + 1] * k_s[llo * LDPAD + kb + 1];
            a1[0] = g_s[kb]     * k_s[(16 + llo) * LDPAD + kb];
            a1[1] = g_s[kb + 1] * k_s[(16 + llo) * LDPAD + kb + 1];
            const size_t wb = (size_t)(kk >> 2) * (DIM * 4) + 2 * lhi;
            const v2f ba = *(const v2f*)&wpp[wb + (size_t)ca * 4];
            const v2f bb = *(const v2f*)&wpp[wb + (size_t)cb * 4];
            o00 = WMMA_F32(a0, ba, o00);  o10 = WMMA_F32(a1, ba, o10);
            o01 = WMMA_F32(a0, bb, o01);  o11 = WMMA_F32(a1, bb, o11);
        }
        const float bpa = b_proj[ca], gma = gamma[ca], bta = beta[ca];
        const float bpb = b_proj[cb], gmb = gamma[cb], btb = beta[cb];
#pragma unroll
        for (int r = 0; r < 8; ++r) {
            const int r0 = r + 8 * lhi, r1 = 16 + r0;
            const float kva = k_s[r0 * LDPAD + ca], kvb = k_s[r1 * LDPAD + ca];
            const float kvc = k_s[r0 * LDPAD + cb], kvd = k_s[r1 * LDPAD + cb];
            float* o0 = out + ((size_t)b * NPATCH + m0 + r0) * DIM;
            float* o1 = out + ((size_t)b * NPATCH + m0 + r1) * DIM;
            o0[ca] = o00[r] + bpa + (kva - mu[r0]) * rs[r0] * gma + bta;
            o1[ca] = o10[r] + bpa + (kvb - mu[r1]) * rs[r1] * gma + bta;
            o0[cb] = o01[r] + bpb + (kvc - mu[r0]) * rs[r0] * gmb + btb;
            o1[cb] = o11[r] + bpb + (kvd - mu[r1]) * rs[r1] * gmb + btb;
        }
    }
}

// ---------------------------------------------------------------------------
extern "C" void kernel_launch(void* const* d_in, const int* in_sizes, int n_in,
                              void* d_out, int out_size, void* d_ws, size_t ws_size,
                              hipStream_t stream)
{
    (void)in_sizes; (void)n_in; (void)out_size; (void)ws_size;
    const float* x      = (const float*)d_in[0];
    const float* wq     = (const float*)d_in[1];
    const float* bq     = (const float*)d_in[2];
    const float* wk     = (const float*)d_in[3];
    const float* bk     = (const float*)d_in[4];
    const float* w_g    = (const float*)d_in[5];
    const float* w_fc   = (const float*)d_in[6];
    const float* b_fc   = (const float*)d_in[7];
    const float* w_fc2  = (const float*)d_in[8];
    const float* b_fc2  = (const float*)d_in[9];
    const float* w_proj = (const float*)d_in[10];
    const float* b_proj = (const float*)d_in[11];
    const float* gamma  = (const float*)d_in[12];
    const float* beta   = (const float*)d_in[13];
    float* out = (float*)d_out;

    const int BS = 16;
    const size_t WSZ = (size_t)DIM * DIM;       // 589824 floats per weight
    float* ws     = (float*)d_ws;
    float* qw_buf = ws;                         // 16*768
    float* qs_buf = ws + 1 * BS * DIM;
    float* ks_buf = ws + 2 * BS * DIM;
    float* g_buf  = ws + 3 * BS * DIM;
    float* wqp    = ws + 4 * BS * DIM;          // packed weights (3 x 2.25 MB)
    float* wkp    = wqp + WSZ;
    float* wpp    = wkp + WSZ;

    // zero the atomic accumulation buffers (idempotent, graph-capturable)
    hipMemsetAsync(d_ws, 0, (size_t)3 * BS * DIM * sizeof(float), stream);

    const dim3 blk(256);
    const dim3 grdP(DIM * DIM / 256);
    ehia_repack<<<grdP, blk, 0, stream>>>(wq, wqp);
    ehia_repack<<<grdP, blk, 0, stream>>>(wk, wkp);
    ehia_repack<<<grdP, blk, 0, stream>>>(w_proj, wpp);

    const dim3 grdA(NPATCH / MT, BS);
    const size_t ldsA = (size_t)MT * LDPAD * sizeof(float);
    ehia_qk_reduce<<<grdA, blk, ldsA, stream>>>(x, wqp, bq, wkp, bk, w_g,
                                                qw_buf, qs_buf, ks_buf);

    ehia_gate<<<dim3(BS), blk, 0, stream>>>(qw_buf, qs_buf, ks_buf,
                                            w_fc, b_fc, w_fc2, b_fc2, g_buf);

    const dim3 grdC(NPATCH / MT, BS);
    const size_t ldsC = ((size_t)2 * MT * LDPAD + DIM + 2 * MT) * sizeof(float);
    ehia_out<<<grdC, blk, ldsC, stream>>>(x, wkp, bk, g_buf, wpp, b_proj,
                                          gamma, beta, out);
}